// GConv_8134668059125
// MI455X (gfx1250) — compile-verified
//
#include <hip/hip_runtime.h>
#include <hip/hip_bf16.h>

typedef float v2f __attribute__((ext_vector_type(2)));
typedef float v8f __attribute__((ext_vector_type(8)));

#define HDIM 128

// ---------------- small utility kernels ----------------

__global__ void zero_f32(float* p, int n) {
    int i = blockIdx.x * blockDim.x + threadIdx.x;
    if (i < n) p[i] = 0.0f;
}

__global__ void deg_init(float* deg, int n) {
    int i = blockIdx.x * blockDim.x + threadIdx.x;
    if (i < n) deg[i] = 1.0f;   // self-loop contributes 1 to every node's degree
}

__global__ void deg_edges(const int* __restrict__ dst, float* deg, int E) {
    int e = blockIdx.x * blockDim.x + threadIdx.x;
    if (e < E) atomicAdd(&deg[dst[e]], 1.0f);
}

__global__ void deg_finish(float* deg_dinv, int n) {
    int i = blockIdx.x * blockDim.x + threadIdx.x;
    if (i < n) {
        float d = deg_dinv[i];
        deg_dinv[i] = (d > 0.0f) ? rsqrtf(d) : 0.0f;  // d >= 1 always (self-loop)
    }
}

// ---------------- fp32 WMMA GEMM: ZW = Z @ W  (Z:[n,128], W:[128,128]) ----------------
// 8 waves per block; each wave produces a 16x128 output stripe via
// V_WMMA_F32_16X16X4_F32 (32 K-steps x 8 N-tiles).

__global__ __launch_bounds__(256) void gemm_wmma_f32(
        const float* __restrict__ Z, const float* __restrict__ W,
        float* __restrict__ ZW, int nrows) {
    __shared__ float sW[HDIM * HDIM];           // 64 KB: full 128x128 W
    for (int i = threadIdx.x; i < HDIM * HDIM; i += blockDim.x) sW[i] = W[i];
    __syncthreads();

    const int wave = threadIdx.x >> 5;
    const int lane = threadIdx.x & 31;
    const int tile = blockIdx.x * 8 + wave;     // 16-row tile index
    const int m0 = tile * 16;
    if (m0 >= nrows) return;                    // wave-uniform: EXEC stays all-ones

    const int half = lane >> 4;                 // 0: lanes 0-15, 1: lanes 16-31
    const int l16  = lane & 15;

    v8f acc[8] = {};                            // 8 N-tiles of 16 cols each

    const float* zrow = Z + (size_t)(m0 + l16) * HDIM;
    for (int k0 = 0; k0 < HDIM; k0 += 4) {
        const int ka = k0 + half * 2;
        // A 16x4 f32 layout: VGPR0 = K=k0 (lanes0-15) / k0+2 (lanes16-31), VGPR1 = K+1
        v2f afrag;
        afrag.x = zrow[ka];
        afrag.y = zrow[ka + 1];
        #pragma unroll
        for (int t = 0; t < 8; ++t) {
            const int n0 = t * 16;
            // B 4x16 f32 layout mirrors A: VGPR0 = row K=k0/k0+2, VGPR1 = row K+1
            v2f bfrag;
            bfrag.x = sW[ka * HDIM + n0 + l16];
            bfrag.y = sW[(ka + 1) * HDIM + n0 + l16];
            acc[t] = __builtin_amdgcn_wmma_f32_16x16x4_f32(
                false, afrag, false, bfrag, (short)0, acc[t], false, false);
        }
    }

    // C/D 16x16 f32 layout: VGPR r holds row m0+r (lanes0-15) / m0+8+r (lanes16-31)
    const int mbase = m0 + half * 8;
    #pragma unroll
    for (int t = 0; t < 8; ++t) {
        float* outp = ZW + (size_t)mbase * HDIM + t * 16 + l16;
        #pragma unroll
        for (int r = 0; r < 8; ++r) outp[(size_t)r * HDIM] = acc[t][r];
    }
}

// ---------------- aggregation ----------------

// agg[i] = dinv[i]^2 * zw[i]   (self-loop term), vectorized float4
__global__ void agg_init(const float* __restrict__ zw, const float* __restrict__ dinv,
                         float* __restrict__ agg, int n) {
    int i = blockIdx.x * blockDim.x + threadIdx.x;   // over n*32 float4s
    if (i >= n * (HDIM / 4)) return;
    int node = i >> 5;
    float di = dinv[node];
    float w = di * di;
    float4 v = ((const float4*)zw)[i];
    v.x *= w; v.y *= w; v.z *= w; v.w *= w;
    ((float4*)agg)[i] = v;
}

// one wave per edge: agg[dst] += dinv[src]*dinv[dst] * zw[src]
__global__ __launch_bounds__(256) void edge_scatter(
        const float* __restrict__ zw, const float* __restrict__ dinv,
        const int* __restrict__ src, const int* __restrict__ dst,
        float* __restrict__ agg, int E) {
    int e = blockIdx.x * 8 + (threadIdx.x >> 5);
    int lane = threadIdx.x & 31;
    if (e >= E) return;                         // wave-uniform
    int s = src[e], d = dst[e];
    float w = dinv[s] * dinv[d];
    float4 v = ((const float4*)(zw + (size_t)s * HDIM))[lane];
    float* ap = agg + (size_t)d * HDIM + lane * 4;
    atomicAdd(ap + 0, w * v.x);
    atomicAdd(ap + 1, w * v.y);
    atomicAdd(ap + 2, w * v.z);
    atomicAdd(ap + 3, w * v.w);
}

// z = PReLU(agg + b) in place, and pool: g[batch[i]*(2H) + col_off + h] += z[i][h]
__global__ __launch_bounds__(256) void bias_prelu_pool(
        float* __restrict__ z, const float* __restrict__ b,
        const float* __restrict__ alpha, const int* __restrict__ batch,
        float* __restrict__ g, int col_off, int n) {
    int node = blockIdx.x * 8 + (threadIdx.x >> 5);
    int lane = threadIdx.x & 31;
    if (node >= n) return;                      // wave-uniform
    int h = lane * 4;
    float4 v  = *(float4*)(z + (size_t)node * HDIM + h);
    float4 bb = *(const float4*)(b + h);
    float4 aa = *(const float4*)(alpha + h);
    v.x += bb.x; v.y += bb.y; v.z += bb.z; v.w += bb.w;
    v.x = (v.x > 0.0f) ? v.x : aa.x * v.x;
    v.y = (v.y > 0.0f) ? v.y : aa.y * v.y;
    v.z = (v.z > 0.0f) ? v.z : aa.z * v.z;
    v.w = (v.w > 0.0f) ? v.w : aa.w * v.w;
    *(float4*)(z + (size_t)node * HDIM + h) = v;
    float* gp = g + (size_t)batch[node] * (2 * HDIM) + col_off + h;
    atomicAdd(gp + 0, v.x);
    atomicAdd(gp + 1, v.y);
    atomicAdd(gp + 2, v.z);
    atomicAdd(gp + 3, v.w);
}

// ---------------- launcher ----------------

extern "C" void kernel_launch(void* const* d_in, const int* in_sizes, int n_in,
                              void* d_out, int out_size, void* d_ws, size_t ws_size,
                              hipStream_t stream) {
    const float* x     = (const float*)d_in[0];
    const int*   eidx  = (const int*)  d_in[1];
    const int*   batch = (const int*)  d_in[2];
    const float* W1    = (const float*)d_in[3];
    const float* b1    = (const float*)d_in[4];
    const float* W2    = (const float*)d_in[5];
    const float* b2    = (const float*)d_in[6];
    const float* alpha = (const float*)d_in[7];

    const int N = in_sizes[0] / HDIM;        // 50000
    const int E = in_sizes[1] / 2;           // 800000
    const int* src = eidx;
    const int* dst = eidx + E;

    float* ws   = (float*)d_ws;
    float* dinv = ws;                        // [N]
    float* zw   = ws + N;                    // [N*128]  (N*4 bytes is 16B-aligned)
    float* z1   = zw + (size_t)N * HDIM;     // [N*128]

    float* z2 = (float*)d_out;               // [N*128] final layer output
    float* g  = z2 + (size_t)N * HDIM;       // [G*256] pooled output
    const int gElems = out_size - N * HDIM;

    const int T = 256;
    const int nodeBlocks = (N + T - 1) / T;
    const int edgeBlocks = (E + T - 1) / T;
    const int gemmBlocks = ((N + 15) / 16 + 7) / 8;
    const int vecBlocks  = (N * (HDIM / 4) + T - 1) / T;
    const int waveNodeBlocks = (N + 7) / 8;
    const int waveEdgeBlocks = (E + 7) / 8;

    // pooled output must start at zero (harness poisons d_out)
    zero_f32<<<(gElems + T - 1) / T, T, 0, stream>>>(g, gElems);

    // degree -> dinv
    deg_init  <<<nodeBlocks, T, 0, stream>>>(dinv, N);
    deg_edges <<<edgeBlocks, T, 0, stream>>>(dst, dinv, E);
    deg_finish<<<nodeBlocks, T, 0, stream>>>(dinv, N);

    // ---- layer 1: z1 = PReLU(D^-1/2 (A+I) D^-1/2 (x@W1) + b1), pool into g[:, 0:128]
    gemm_wmma_f32  <<<gemmBlocks, T, 0, stream>>>(x, W1, zw, N);
    agg_init       <<<vecBlocks, T, 0, stream>>>(zw, dinv, z1, N);
    edge_scatter   <<<waveEdgeBlocks, T, 0, stream>>>(zw, dinv, src, dst, z1, E);
    bias_prelu_pool<<<waveNodeBlocks, T, 0, stream>>>(z1, b1, alpha, batch, g, 0, N);

    // ---- layer 2: z2 = PReLU(... (z1@W2) + b2), pool into g[:, 128:256]
    gemm_wmma_f32  <<<gemmBlocks, T, 0, stream>>>(z1, W2, zw, N);
    agg_init       <<<vecBlocks, T, 0, stream>>>(zw, dinv, z2, N);
    edge_scatter   <<<waveEdgeBlocks, T, 0, stream>>>(zw, dinv, src, dst, z2, E);
    bias_prelu_pool<<<waveNodeBlocks, T, 0, stream>>>(z2, b2, alpha, batch, g, HDIM, N);
}